// WeightedAverageLayer_13357348291340
// MI455X (gfx1250) — compile-verified
//
#include <hip/hip_runtime.h>
#include <cstdint>

// ---------------- configuration ----------------
#define D               128        // feature dim (fixed by reference)
#define TILE_ROWS       32         // rows per LDS tile
#define ROWS_PER_BLOCK  512        // rows per workgroup
#define TILES_PER_BLOCK (ROWS_PER_BLOCK / TILE_ROWS)
#define SEG_TILE        16         // segments per WMMA tile

typedef __attribute__((ext_vector_type(2))) float v2f;
typedef __attribute__((ext_vector_type(8))) float v8f;

// ---------------- CDNA5 async global->LDS helpers ----------------
// VGLOBAL async copy (GV mode): VDST = LDS byte offset, VADDR = 64-bit global
// address. Tracked by ASYNCcnt. NT hint: 512MB stream, don't pollute L2.
__device__ __forceinline__ void async_b128_to_lds(uint32_t lds_byte_off,
                                                  const float* gptr) {
  asm volatile("global_load_async_to_lds_b128 %0, %1, off th:TH_LOAD_NT"
               :: "v"(lds_byte_off),
                  "v"((unsigned long long)(uintptr_t)gptr)
               : "memory");
}
__device__ __forceinline__ void wait_asynccnt_le8() {
  asm volatile("s_wait_asynccnt 8" ::: "memory");   // tile t done, t+1 in flight
}
__device__ __forceinline__ void wait_asynccnt_0() {
  asm volatile("s_wait_asynccnt 0" ::: "memory");
}

// ---------------- kernel 1: segmented weighted sums ----------------
// 128 threads (4 waves); thread owns one of 128 columns. Double-buffered
// async LDS streaming of prop; run-length flush with f32 global atomics.
__global__ __launch_bounds__(128) void segsum_kernel(
    const float* __restrict__ prop, const float* __restrict__ weights,
    const int* __restrict__ index, float* __restrict__ psum,
    float* __restrict__ wsum, int n)
{
  extern __shared__ float lds[];   // 2 * TILE_ROWS * D floats = 32 KB
  const int tid  = threadIdx.x;    // column 0..127
  const int lane = tid & 31;
  const long chunk0 = (long)blockIdx.x * ROWS_PER_BLOCK;
  if (chunk0 >= n) return;
  const long rows_left = (long)n - chunk0;
  const int rows_here  = rows_left < ROWS_PER_BLOCK ? (int)rows_left : ROWS_PER_BLOCK;
  const int tiles      = (rows_here + TILE_ROWS - 1) / TILE_ROWS;

  auto issue_tile = [&](int t, int buf) {
    const long row0 = chunk0 + (long)t * TILE_ROWS;
    // exactly 8 async instructions per wave per tile (addresses clamped, so
    // the per-wave ASYNCcnt arithmetic in the pipeline is invariant)
#pragma unroll
    for (int k = 0; k < 8; ++k) {
      const int chunk = k * 128 + tid;   // 0..1023 b128 chunks of the tile
      const int elem  = chunk * 4;       // float index within tile
      const int r     = elem >> 7;       // elem / 128
      const int c     = elem & 127;
      long grow = row0 + r;
      if (grow > (long)n - 1) grow = (long)n - 1;     // clamp OOB (unused data)
      const uint32_t lo = (uint32_t)buf * (TILE_ROWS * D * 4u) + (uint32_t)elem * 4u;
      async_b128_to_lds(lo, prop + grow * D + c);
    }
  };

  issue_tile(0, 0);

  int   cur_seg = index[chunk0];
  float acc  = 0.0f;   // weighted column sum for current run
  float wacc = 0.0f;   // weight sum for current run (uniform across threads)

  for (int t = 0; t < tiles; ++t) {
    const int buf = t & 1;
    if (t + 1 < tiles) { issue_tile(t + 1, buf ^ 1); wait_asynccnt_le8(); }
    else               { wait_asynccnt_0(); }
    __syncthreads();   // all waves' tile-t data resident in LDS

    const long row0 = chunk0 + (long)t * TILE_ROWS;
    long mr = row0 + lane; if (mr > (long)n - 1) mr = (long)n - 1;
    const int   myidx = index[mr];     // lane l caches row (row0+l) metadata
    const float myw   = weights[mr];
    const long  rem   = (long)n - row0;
    const int   rows  = rem < TILE_ROWS ? (int)rem : TILE_ROWS;
    const float* lbuf = lds + buf * (TILE_ROWS * D);

    for (int r = 0; r < rows; ++r) {
      const int   seg = __shfl(myidx, r, 32);   // broadcast row metadata
      const float w   = __shfl(myw,   r, 32);
      if (seg != cur_seg) {                     // run boundary: flush
        unsafeAtomicAdd(&psum[(long)cur_seg * D + tid], acc);
        if (tid == 0) unsafeAtomicAdd(&wsum[cur_seg], wacc);
        acc = 0.0f; wacc = 0.0f; cur_seg = seg;
      }
      acc   = fmaf(w, lbuf[r * D + tid], acc);
      wacc += w;
    }
    __syncthreads();   // everyone done reading buf before it is overwritten
  }
  unsafeAtomicAdd(&psum[(long)cur_seg * D + tid], acc);
  if (tid == 0) unsafeAtomicAdd(&wsum[cur_seg], wacc);
}

// ---------------- kernel 2: normalization via WMMA ----------------
// out[s][:] *= 1/wsum[s]  ==  diag(1/wsum) x psum, done per 16x16 tile with
// four chained v_wmma_f32_16x16x4_f32 (K=4 slices of the 16x16 diagonal).
// Block = 256 threads (8 waves); wave w handles column tile [16w,16w+16).
__global__ __launch_bounds__(256) void scale_wmma_kernel(
    float* __restrict__ out, const float* __restrict__ wsum, int S)
{
  const int wave = threadIdx.x >> 5;
  const int lane = threadIdx.x & 31;
  const int s0   = blockIdx.x * SEG_TILE;
  if (s0 >= S) return;
  const int c0   = wave * 16;

  const int sl     = lane & 15;                  // M row / N col slot
  const int sclamp = (s0 + sl < S) ? (s0 + sl) : (S - 1);
  const float inv  = 1.0f / wsum[sclamp];        // empty seg -> inf -> NaN (matches 0/0)
  const int kbase  = (lane >> 4) * 2;            // K pair held by this half-wave

  v8f c = {};
#pragma unroll
  for (int j = 0; j < 4; ++j) {
    // A (16x4 f32): lane<16 holds A[sl][0..1], lane>=16 holds A[sl][2..3]
    v2f a;
    a.x = (sl == 4 * j + kbase)     ? inv : 0.0f;
    a.y = (sl == 4 * j + kbase + 1) ? inv : 0.0f;
    // B (4x16 f32): rows are the same 16 segments (diagonal product)
    int r0 = s0 + 4 * j + kbase;     if (r0 > S - 1) r0 = S - 1;
    int r1 = s0 + 4 * j + kbase + 1; if (r1 > S - 1) r1 = S - 1;
    v2f b;
    b.x = out[(long)r0 * D + c0 + sl];
    b.y = out[(long)r1 * D + c0 + sl];
    c = __builtin_amdgcn_wmma_f32_16x16x4_f32(
        /*neg_a=*/false, a, /*neg_b=*/false, b,
        /*c_mod=*/(short)0, c, /*reuse_a=*/false, /*reuse_b=*/false);
  }
  // C/D layout: VGPR v -> M = v + 8*(lane>>4), N = sl
#pragma unroll
  for (int v = 0; v < 8; ++v) {
    const int m = v + 8 * (lane >> 4);
    if (s0 + m < S) out[(long)(s0 + m) * D + c0 + sl] = c[v];
  }
}

// ---------------- host launcher ----------------
extern "C" void kernel_launch(void* const* d_in, const int* in_sizes, int n_in,
                              void* d_out, int out_size, void* d_ws, size_t ws_size,
                              hipStream_t stream) {
  const float* prop    = (const float*)d_in[0];
  const float* weights = (const float*)d_in[1];
  const int*   index   = (const int*)d_in[2];
  float* out  = (float*)d_out;
  float* wsum = (float*)d_ws;                 // S floats of scratch
  const int n = in_sizes[1];                  // 1,000,000 rows
  const int S = out_size / D;                 // 50,000 segments

  hipMemsetAsync(d_out, 0, (size_t)out_size * sizeof(float), stream);
  hipMemsetAsync(d_ws,  0, (size_t)S * sizeof(float), stream);

  const int nblocks = (n + ROWS_PER_BLOCK - 1) / ROWS_PER_BLOCK;
  segsum_kernel<<<nblocks, 128, 2 * TILE_ROWS * D * sizeof(float), stream>>>(
      prop, weights, index, out, wsum, n);

  const int sblocks = (S + SEG_TILE - 1) / SEG_TILE;
  scale_wmma_kernel<<<sblocks, 256, 0, stream>>>(out, wsum, S);
}